// SolveSchedulingQP_41850161332249
// MI455X (gfx1250) — compile-verified
//
#include <hip/hip_runtime.h>

// Batched QP (ramp-constrained) primal-dual IPM, exploiting the tridiagonal
// structure of G' diag(w) G + diag(q) for G = [D; -D] (first differences).
// One wave32 per batch; PCR tridiagonal solve via ds_bpermute; inputs staged
// through the CDNA5 Tensor Data Mover (tensor_load_to_lds + s_wait_tensorcnt).

#define NVAR 64
#define NITERS 30
#define SIGMA_C 0.1f
#define CRAMP 0.4f
#define BIGV 1e12f
#define WAVES_PER_BLOCK 8
#define BLOCK_THREADS 256

typedef unsigned int u32x4 __attribute__((ext_vector_type(4)));
typedef int i32x4 __attribute__((ext_vector_type(4)));
typedef int i32x8 __attribute__((ext_vector_type(8)));

// ---------- wave32 cross-lane helpers (ds_bpermute; index wraps mod 32) ----
__device__ __forceinline__ float lanef(float v, int srcLane) {
  return __int_as_float(__builtin_amdgcn_ds_bpermute(srcLane << 2, __float_as_int(v)));
}

__device__ __forceinline__ float frcp(float x) {
#if __has_builtin(__builtin_amdgcn_rcpf)
  return __builtin_amdgcn_rcpf(x);   // v_rcp_f32: ~1ulp, fine inside an IPM
#else
  return 1.0f / x;
#endif
}

__device__ __forceinline__ float wave_sum(float v, int lane) {
#pragma unroll
  for (int o = 16; o > 0; o >>= 1) v += lanef(v, lane ^ o);
  return v;  // all lanes hold the total
}

__device__ __forceinline__ float wave_min(float v, int lane) {
#pragma unroll
  for (int o = 16; o > 0; o >>= 1) v = fminf(v, lanef(v, lane ^ o));
  return v;
}

// ---------- PCR row ops -----------------------------------------------------
struct R4 { float a, b, c, d; };  // a: sub-diag, b: diag, c: super-diag, d: rhs

__device__ __forceinline__ R4 row_gather(R4 r, int src, bool inval) {
  // bpermute executed by all lanes (EXEC full); identity row selected after.
  float ga = lanef(r.a, src), gb = lanef(r.b, src),
        gc = lanef(r.c, src), gd = lanef(r.d, src);
  R4 o;
  o.a = inval ? 0.0f : ga;
  o.b = inval ? 1.0f : gb;
  o.c = inval ? 0.0f : gc;
  o.d = inval ? 0.0f : gd;
  return o;
}

__device__ __forceinline__ R4 pcr_upd(R4 r, R4 L, R4 Rt) {
  float al = -r.a * frcp(L.b);
  float ga = -r.c * frcp(Rt.b);
  R4 o;
  o.a = al * L.a;
  o.c = ga * Rt.c;
  o.b = r.b + al * L.c + ga * Rt.a;
  o.d = r.d + al * L.d + ga * Rt.d;
  return o;
}

// ---------- Tensor Data Mover staging (CDNA5 async-tensor path) ------------
#if __has_builtin(__builtin_amdgcn_tensor_load_to_lds)
#define HAVE_TDM 1
#else
#define HAVE_TDM 0
#endif

#if HAVE_TDM
__device__ __forceinline__ void wait_tensor0() {
#if __has_builtin(__builtin_amdgcn_s_wait_tensorcnt)
  __builtin_amdgcn_s_wait_tensorcnt(0);
#else
  asm volatile("s_wait_tensorcnt 0x0" ::: "memory");
#endif
}

// 1-D contiguous copy of nelem f32 from global to LDS via one D# descriptor.
// Group0: count=1, lds_addr, global_addr[56:0], type=2.
// Group1: data_size=4B, tensor_dim0=nelem, tensor_dim1=1, tile_dim0=nelem,
//         tile_dim1=1, tensor_dim0_stride=nelem, tensor_dim1_stride=nelem.
__device__ __forceinline__ void tdm_load_1d(const void* gsrc,
                                            unsigned lds_byte_off,
                                            unsigned nelem) {
  unsigned long long ga = (unsigned long long)(uintptr_t)gsrc;
  u32x4 g0 = { 1u,                                   // count=1, user mode
               lds_byte_off,                         // lds_addr
               (unsigned)(ga & 0xFFFFFFFFull),       // global_addr lo
               (unsigned)((ga >> 32) & 0x1FFFFFFull) // global_addr hi (56:32)
                   | 0x80000000u };                  // type=2 at [127:126]
  i32x8 g1;
  g1[0] = 0x00020000;                                   // data_size=2 (4B)
  g1[1] = (int)((nelem & 0xFFFFu) << 16);               // tensor_dim0 lo16
  g1[2] = (int)(((nelem >> 16) & 0xFFFFu) | (1u << 16)); // dim0 hi | tensor_dim1=1
  g1[3] = (int)((nelem & 0xFFFFu) << 16);               // tile_dim0 (<=512)
  g1[4] = 1;                                            // tile_dim1=1, tile_dim2=0
  g1[5] = (int)nelem;                                   // tensor_dim0_stride lo32
  g1[6] = (int)((nelem & 0xFFFFu) << 16);               // dim1_stride lo16
  g1[7] = (int)((nelem >> 16) & 0xFFFFu);               // dim1_stride hi
  i32x4 z4 = {0, 0, 0, 0};
#if defined(__clang_major__) && (__clang_major__ >= 23)
  i32x8 z8 = {0, 0, 0, 0, 0, 0, 0, 0};
  __builtin_amdgcn_tensor_load_to_lds(g0, g1, z4, z4, z8, 0);
#else
  __builtin_amdgcn_tensor_load_to_lds(g0, g1, z4, z4, 0);
#endif
}
#endif  // HAVE_TDM

// ---------- kernel ----------------------------------------------------------
__global__ __launch_bounds__(BLOCK_THREADS) void qp_tridiag_ipm_kernel(
    const float* __restrict__ z0g, const float* __restrict__ mug,
    const float* __restrict__ dgg, const float* __restrict__ d2gg,
    float* __restrict__ out, int B) {
  __shared__ float S[4][WAVES_PER_BLOCK * NVAR];

  const int tid = (int)threadIdx.x;
  const int wid = tid >> 5;
  const int lane = tid & 31;
  const int bbase = (int)blockIdx.x * WAVES_PER_BLOCK;
  const long long goff = (long long)bbase * NVAR;
  int remf = (B - bbase) * NVAR;
  const unsigned nelem =
      (unsigned)(remf < WAVES_PER_BLOCK * NVAR ? remf : WAVES_PER_BLOCK * NVAR);

#if HAVE_TDM
  if (wid == 0) {
    unsigned s0 = (unsigned)(unsigned long long)(uintptr_t)(&S[0][0]);
    const unsigned span = (unsigned)(sizeof(float) * WAVES_PER_BLOCK * NVAR);
    tdm_load_1d(z0g + goff,  s0 + 0u * span, nelem);
    tdm_load_1d(mug + goff,  s0 + 1u * span, nelem);
    tdm_load_1d(dgg + goff,  s0 + 2u * span, nelem);
    tdm_load_1d(d2gg + goff, s0 + 3u * span, nelem);
    wait_tensor0();
  }
#else
  for (unsigned i = (unsigned)tid; i < nelem; i += BLOCK_THREADS) {
    S[0][i] = z0g[goff + i];
    S[1][i] = mug[goff + i];
    S[2][i] = dgg[goff + i];
    S[3][i] = d2gg[goff + i];
  }
#endif
  __syncthreads();

  const int b = bbase + wid;
  if (b >= B) return;  // no barriers past this point

  // Lane L owns variables j = 2L, 2L+1 and constraint pairs i = 2L, 2L+1.
  const int base = wid * NVAR + 2 * lane;
  const bool v1 = (lane < 31);  // slot-b constraint (i=2L+1<=62) valid
  const bool l0 = (lane == 0);
  const bool l31 = (lane == 31);

  float z0a = S[0][base], z0b = S[0][base + 1];
  float mua = S[1][base], mub = S[1][base + 1];
  float dga = S[2][base], dgb = S[2][base + 1];
  float d2a = S[3][base], d2b = S[3][base + 1];

  float qa = d2a + 1.0f, qb = d2b + 1.0f;
  float pa = dga - d2a * z0a - mua;
  float pb = dgb - d2b * z0b - mub;

  float za = 0.0f, zb = 0.0f;
  float lDa = 1.0f, lDb = 1.0f, lNa = 1.0f, lNb = 1.0f;
  float sDa = CRAMP, sDb = CRAMP, sNa = CRAMP, sNb = CRAMP;

#pragma unroll 1
  for (int it = 0; it < NITERS; ++it) {
    // ---- u = D z;  primal residuals r_p = Gz + s - h -----------------------
    float zaR = lanef(za, lane + 1);  // z[2L+2] (garbage at lane31, masked)
    float ua = za - zb;               // u[2L]
    float ub = zb - zaR;              // u[2L+1] (invalid at lane31)
    float rpDa = ua + sDa - CRAMP, rpDb = ub + sDb - CRAMP;
    float rpNa = -ua + sNa - CRAMP, rpNb = -ub + sNb - CRAMP;

    // ---- duality gap mu = mean(s*lam) over m=126 ---------------------------
    float acc = sDa * lDa + sNa * lNa + (v1 ? (sDb * lDb + sNb * lNb) : 0.0f);
    float mugap = wave_sum(acc, lane) * (1.0f / 126.0f);
    float smu = SIGMA_C * mugap;

    // ---- w = lam / s (masked on nonexistent constraint slot) ---------------
    float wDa = lDa * frcp(sDa), wNa = lNa * frcp(sNa);
    float wDb = v1 ? lDb * frcp(sDb) : 0.0f;
    float wNb = v1 ? lNb * frcp(sNb) : 0.0f;

    // ---- dual residual r_d = q*z + p + (lamD-lamN)@D -----------------------
    float ta = lDa - lNa;
    float tb = v1 ? (lDb - lNb) : 0.0f;  // t[63]:=0 convention
    float tbL = lanef(tb, lane - 1);
    tbL = l0 ? 0.0f : tbL;               // t[-1]:=0
    float rda = qa * za + pa + (ta - tbL);
    float rdb = qb * zb + pb + (tb - ta);

    // ---- rhs = -(r_d + (gD-gN)@D), g = (smu/s - lam) + w*r_p ---------------
    float ea = (smu * frcp(sDa) - lDa) + wDa * rpDa -
               ((smu * frcp(sNa) - lNa) + wNa * rpNa);
    float eb = v1 ? ((smu * frcp(sDb) - lDb) + wDb * rpDb -
                     ((smu * frcp(sNb) - lNb) + wNb * rpNb))
                  : 0.0f;
    float ebL = lanef(eb, lane - 1);
    ebL = l0 ? 0.0f : ebL;
    float rhsa = -(rda + ea - ebL);
    float rhsb = -(rdb + eb - ea);

    // ---- tridiagonal M: diag = q + c[j] + c[j-1], off[j] = -c[j] -----------
    float ca = wDa + wNa;
    float cb = wDb + wNb;  // 0 at lane31 (c[63]:=0)
    float cbL = lanef(cb, lane - 1);
    cbL = l0 ? 0.0f : cbL;
    R4 r0{-cbL, qa + ca + cbL, -ca, rhsa};  // row 2L
    R4 r1{-ca, qb + cb + ca, -cb, rhsb};    // row 2L+1

    // ---- PCR: 6 steps over 64 rows (2 rows/lane) ---------------------------
    {  // s = 1: cross-slot neighbors
      R4 Lm = row_gather(r1, lane - 1, l0);   // row 2L-1
      R4 Rp = row_gather(r0, lane + 1, l31);  // row 2L+2
      R4 n0 = pcr_upd(r0, Lm, r1);
      R4 n1 = pcr_upd(r1, r0, Rp);
      r0 = n0;
      r1 = n1;
    }
#pragma unroll
    for (int t = 1; t <= 16; t <<= 1) {  // s = 2t: slot-preserving
      bool lo = lane < t, hi = lane >= 32 - t;
      R4 L0 = row_gather(r0, lane - t, lo);
      R4 R0 = row_gather(r0, lane + t, hi);
      R4 L1 = row_gather(r1, lane - t, lo);
      R4 R1 = row_gather(r1, lane + t, hi);
      r0 = pcr_upd(r0, L0, R0);
      r1 = pcr_upd(r1, L1, R1);
    }
    float dza = r0.d * frcp(r0.b);
    float dzb = r1.d * frcp(r1.b);

    // ---- ds, dlam ----------------------------------------------------------
    float dzaR = lanef(dza, lane + 1);
    float dua = dza - dzb;
    float dub = dzb - dzaR;
    float dsDa = -rpDa - dua, dsNa = -rpNa + dua;
    float dsDb = v1 ? (-rpDb - dub) : 0.0f;
    float dsNb = v1 ? (-rpNb + dub) : 0.0f;
    float dlDa = (smu - lDa * (sDa + dsDa)) * frcp(sDa);
    float dlNa = (smu - lNa * (sNa + dsNa)) * frcp(sNa);
    float dlDb = v1 ? (smu - lDb * (sDb + dsDb)) * frcp(sDb) : 0.0f;
    float dlNb = v1 ? (smu - lNb * (sNb + dsNb)) * frcp(sNb) : 0.0f;

    // ---- fraction-to-boundary step -----------------------------------------
    float r = BIGV;
    r = fminf(r, (dsDa < 0.0f) ? (-sDa * frcp(dsDa)) : BIGV);
    r = fminf(r, (dsNa < 0.0f) ? (-sNa * frcp(dsNa)) : BIGV);
    r = fminf(r, (dsDb < 0.0f) ? (-sDb * frcp(dsDb)) : BIGV);
    r = fminf(r, (dsNb < 0.0f) ? (-sNb * frcp(dsNb)) : BIGV);
    r = fminf(r, (dlDa < 0.0f) ? (-lDa * frcp(dlDa)) : BIGV);
    r = fminf(r, (dlNa < 0.0f) ? (-lNa * frcp(dlNa)) : BIGV);
    r = fminf(r, (dlDb < 0.0f) ? (-lDb * frcp(dlDb)) : BIGV);
    r = fminf(r, (dlNb < 0.0f) ? (-lNb * frcp(dlNb)) : BIGV);
    float alpha = fminf(1.0f, 0.99f * wave_min(r, lane));

    // ---- update ------------------------------------------------------------
    za += alpha * dza;
    zb += alpha * dzb;
    lDa += alpha * dlDa;
    lDb += alpha * dlDb;
    lNa += alpha * dlNa;
    lNb += alpha * dlNb;
    sDa += alpha * dsDa;
    sDb += alpha * dsDb;
    sNa += alpha * dsNa;
    sNb += alpha * dsNb;
  }

  float2 o2;
  o2.x = za;
  o2.y = zb;
  *reinterpret_cast<float2*>(out + goff + (long long)wid * NVAR + 2 * lane) = o2;
}

extern "C" void kernel_launch(void* const* d_in, const int* in_sizes, int n_in,
                              void* d_out, int out_size, void* d_ws,
                              size_t ws_size, hipStream_t stream) {
  (void)n_in;
  (void)out_size;
  (void)d_ws;
  (void)ws_size;
  const float* z0 = (const float*)d_in[0];
  const float* mu = (const float*)d_in[1];
  const float* dg = (const float*)d_in[2];
  const float* d2g = (const float*)d_in[3];
  float* out = (float*)d_out;
  int B = in_sizes[0] / NVAR;  // 4096
  int blocks = (B + WAVES_PER_BLOCK - 1) / WAVES_PER_BLOCK;
  qp_tridiag_ipm_kernel<<<blocks, BLOCK_THREADS, 0, stream>>>(z0, mu, dg, d2g,
                                                              out, B);
}